// SGC_8280696947367
// MI455X (gfx1250) — compile-verified
//
#include <hip/hip_runtime.h>

#define F 128
#define CLS 10

typedef __attribute__((ext_vector_type(2))) float v2f;
typedef __attribute__((ext_vector_type(8))) float v8f;

// ---------------------------------------------------------------- zero scratch
__global__ void sgc_zero_kernel(float* __restrict__ p, long n) {
    long i = (long)blockIdx.x * blockDim.x + threadIdx.x;
    long stride = (long)gridDim.x * blockDim.x;
    for (; i < n; i += stride) p[i] = 0.0f;
}

// ------------------------------------------------- pad W[128,10] -> Wp[128,16]
__global__ void sgc_prepw_kernel(const float* __restrict__ W, float* __restrict__ Wp) {
    int i = blockIdx.x * blockDim.x + threadIdx.x;   // 128*16 = 2048 threads
    if (i >= F * 16) return;
    int k = i >> 4, n = i & 15;
    Wp[i] = (n < CLS) ? W[k * CLS + n] : 0.0f;
}

// ------------------------------------- SpMM: dst[row] += src[col], wave/edge
// One wave32 per edge; lane handles 4 contiguous floats (b128 gather,
// 4x global_atomic_add_f32 scatter; h fits in 192MB L2).
__global__ __launch_bounds__(256) void sgc_spmm_kernel(
    const float* __restrict__ src, const int* __restrict__ ei,
    float* __restrict__ dst, int E) {
    int e = blockIdx.x * (blockDim.x >> 5) + (threadIdx.x >> 5);
    if (e >= E) return;
    int lane = threadIdx.x & 31;
    int row = ei[e];          // edge_index[0][e]
    int col = ei[E + e];      // edge_index[1][e]
    const float4 v = *(const float4*)(src + (long)col * F + lane * 4);
    float* d = dst + (long)row * F + lane * 4;
    unsafeAtomicAdd(d + 0, v.x);
    unsafeAtomicAdd(d + 1, v.y);
    unsafeAtomicAdd(d + 2, v.z);
    unsafeAtomicAdd(d + 3, v.w);
}

// --------------------------------------------- per-graph sums + node counts
__global__ __launch_bounds__(256) void sgc_pool_kernel(
    const float* __restrict__ h, const int* __restrict__ batch,
    float* __restrict__ sums, float* __restrict__ counts, int N) {
    int i = blockIdx.x * (blockDim.x >> 5) + (threadIdx.x >> 5);
    if (i >= N) return;
    int lane = threadIdx.x & 31;
    int g = batch[i];
    const float4 v = *(const float4*)(h + (long)i * F + lane * 4);
    float* s = sums + (long)g * F + lane * 4;
    unsafeAtomicAdd(s + 0, v.x);
    unsafeAtomicAdd(s + 1, v.y);
    unsafeAtomicAdd(s + 2, v.z);
    unsafeAtomicAdd(s + 3, v.w);
    if (lane == 0) unsafeAtomicAdd(&counts[g], 1.0f);
}

// ------------------------------------------------------- pooled = sums/count
__global__ void sgc_norm_kernel(const float* __restrict__ sums,
                                const float* __restrict__ counts,
                                float* __restrict__ pooled, int GF) {
    int i = blockIdx.x * blockDim.x + threadIdx.x;
    if (i >= GF) return;
    float c = counts[i / F];
    pooled[i] = sums[i] / fmaxf(c, 1.0f);
}

// --------------- final classifier: pooled[G,128] @ Wp[128,16] via fp32 WMMA
// One wave per 16-row tile. A fragment (16x4 f32): lane&15 = M row,
// lane>>4 selects K-pair {k0,k0+1} vs {k0+2,k0+3}; 2 VGPRs = K within pair.
// B fragment (4x16 f32): mirrored — lane&15 = N col, lane half = K-pair.
// C/D (16x16 f32): 8 VGPRs, lanes 0-15 -> M=v, lanes 16-31 -> M=v+8, N=lane&15.
__global__ __launch_bounds__(32) void sgc_gemm_wmma_kernel(
    const float* __restrict__ pooled, const float* __restrict__ Wp,
    const float* __restrict__ bias, float* __restrict__ out) {
    const int tile = blockIdx.x;          // 16-row tile of pooled
    const int lane = threadIdx.x;         // wave32
    const int half = lane >> 4;           // K-pair selector
    const int l15  = lane & 15;
    const int m = tile * 16 + l15;

    v8f acc = {};
    #pragma unroll 4
    for (int k0 = 0; k0 < F; k0 += 4) {
        const int ka = k0 + half * 2;
        v2f a;
        a.x = pooled[(long)m * F + ka];
        a.y = pooled[(long)m * F + ka + 1];
        v2f b;
        b.x = Wp[(ka + 0) * 16 + l15];
        b.y = Wp[(ka + 1) * 16 + l15];
        // D = A(16x4) * B(4x16) + C, full fp32
        acc = __builtin_amdgcn_wmma_f32_16x16x4_f32(
            /*neg_a=*/false, a, /*neg_b=*/false, b,
            /*c_mod=*/(short)0, acc, /*reuse_a=*/false, /*reuse_b=*/false);
    }

    const int n = l15;
    if (n < CLS) {
        const float bb = bias[n];
        const int rbase = tile * 16 + half * 8;
        #pragma unroll
        for (int v = 0; v < 8; ++v)
            out[(rbase + v) * CLS + n] = acc[v] + bb;
    }
}

extern "C" void kernel_launch(void* const* d_in, const int* in_sizes, int n_in,
                              void* d_out, int out_size, void* d_ws, size_t ws_size,
                              hipStream_t stream) {
    const float* x     = (const float*)d_in[0];
    const int*   ei    = (const int*)d_in[1];
    const int*   batch = (const int*)d_in[2];
    const float* W     = (const float*)d_in[3];
    const float* bvec  = (const float*)d_in[4];

    const int N = in_sizes[0] / F;        // 100000
    const int E = in_sizes[1] / 2;        // 1600000
    const int C = in_sizes[4];            // 10
    const int G = out_size / C;           // 256

    // workspace layout (floats), contiguous so one zero pass covers h1..counts
    float* ws = (float*)d_ws;
    const long NF = (long)N * F;
    float* h1     = ws;                   // N*F
    float* h2     = h1 + NF;              // N*F
    float* sums   = h2 + NF;              // G*F
    float* counts = sums + (long)G * F;   // G
    float* pooled = counts + G;           // G*F
    float* Wp     = pooled + (long)G * F; // 128*16

    const long zero_n = 2 * NF + (long)G * F + G;   // h1, h2, sums, counts
    sgc_zero_kernel<<<2048, 256, 0, stream>>>(h1, zero_n);
    sgc_prepw_kernel<<<(F * 16 + 255) / 256, 256, 0, stream>>>(W, Wp);

    const int edges_per_blk = 256 / 32;   // 8 waves per block
    sgc_spmm_kernel<<<(E + edges_per_blk - 1) / edges_per_blk, 256, 0, stream>>>(x,  ei, h1, E);
    sgc_spmm_kernel<<<(E + edges_per_blk - 1) / edges_per_blk, 256, 0, stream>>>(h1, ei, h2, E);

    sgc_pool_kernel<<<(N + edges_per_blk - 1) / edges_per_blk, 256, 0, stream>>>(h2, batch, sums, counts, N);

    const int GF = G * F;
    sgc_norm_kernel<<<(GF + 255) / 256, 256, 0, stream>>>(sums, counts, pooled, GF);

    sgc_gemm_wmma_kernel<<<G / 16, 32, 0, stream>>>(pooled, Wp, bvec, (float*)d_out);
}